// AudioOnlyOnTheFlyModel_44126493999279
// MI455X (gfx1250) — compile-verified
//
#include <hip/hip_runtime.h>
#include <hip/hip_bf16.h>
#include <math.h>

// ---------------------------------------------------------------------------
// Sizes from the reference
// ---------------------------------------------------------------------------
#define LCH     44100          // chirp / rir length
#define USEFUL  2646           // kept conv samples
#define NFREQ   257            // rfft bins of nfft=512
#define NPADF   272            // NFREQ padded to 17*16
#define NFRAMES 166            // 1 + USEFUL/HOP
#define MROWS   176            // NFRAMES padded to 11*16
#define PITCH   65             // LDS row pitch (floats) -> conflict-free
#define NT_CONV 166            // ceil(USEFUL/16) output t-tiles for conv

typedef __attribute__((ext_vector_type(2))) float v2f;
typedef __attribute__((ext_vector_type(8))) float v8f;

// ---------------------------------------------------------------------------
// Kernel A: DFT basis  basis[0][j][f]=cos(2*pi*f*(j+224)/512)
//                      basis[1][j][f]=sin(2*pi*f*(j+224)/512)   (f<257, else 0)
// ---------------------------------------------------------------------------
__global__ void build_basis(float* __restrict__ basis) {
    int idx = blockIdx.x * 256 + threadIdx.x;
    if (idx >= 64 * NPADF) return;
    int j = idx / NPADF;
    int f = idx % NPADF;
    float cv = 0.0f, sv = 0.0f;
    if (f < NFREQ) {
        float ang = 6.283185307179586f * (float)f * (float)(j + 224) * (1.0f / 512.0f);
        __sincosf(ang, &sv, &cv);
    }
    basis[idx]              = cv;   // cos plane
    basis[64 * NPADF + idx] = sv;   // sin plane
}

// ---------------------------------------------------------------------------
// Kernel B: time-domain convolution as Toeplitz GEMM via V_WMMA_F32_16X16X4_F32
//   y[b,c,t] = sum_k rir[b,c,k] * chirp[c, t-k],  t in [0,USEFUL)
// Block = one 16-wide t-tile; 8 waves = (channel c in 0..1) x (batch tile 0..3).
// A fragment (16x4):  a[m][k]   = rir[row(m), k]
// B fragment (4x16):  b[k][n]   = chirp[t(n) - k]   (0 when t<k)
// Main loop (k <= tbase-16) needs no chirp guards: i1 = t-k0-1 >= 14 there.
// ---------------------------------------------------------------------------
__global__ __launch_bounds__(256) void conv_wmma(const float* __restrict__ rir,
                                                 const float* __restrict__ chirp,
                                                 float* __restrict__ y) {
    const int wave  = threadIdx.x >> 5;      // 0..7
    const int lane  = threadIdx.x & 31;
    const int c     = wave >> 2;             // channel
    const int mtile = wave & 3;              // batch tile (16 batches each)
    const int tbase = blockIdx.x * 16;

    const int m    = lane & 15;              // A row within tile
    const int half = lane >> 4;              // selects K pair
    const int n    = lane & 15;              // B/D column within tile
    const int t    = tbase + n;

    const float* __restrict__ rrow = rir + ((size_t)(mtile * 16 + m) * 2 + c) * LCH;
    const float* __restrict__ ch   = chirp + (size_t)c * LCH;

    v8f acc = {0.f, 0.f, 0.f, 0.f, 0.f, 0.f, 0.f, 0.f};

    const int kh    = half * 2;              // K offset for this lane half
    const int kfast = (tbase >= 16) ? (tbase - 16) : 0;  // guard-free bound
    const int kmax  = tbase + 16;            // Toeplitz: k > t contributes 0

    // ---- fast main loop: all chirp indices provably in range ----
    for (int k = 0; k < kfast; k += 4) {
        const int k0 = k + kh;
        v2f a;
        a.x = rrow[k0];
        a.y = rrow[k0 + 1];
        const int i1 = t - k0 - 1;           // >= 14 here
        v2f b;
        b.y = ch[i1];                        // K = k0+1
        b.x = ch[i1 + 1];                    // K = k0
        acc = __builtin_amdgcn_wmma_f32_16x16x4_f32(
                  false, a, false, b, (short)0, acc, false, false);
    }

    // ---- guarded tail: at most 8 iterations near the diagonal ----
    for (int k = kfast; k < kmax; k += 4) {
        const int k0 = k + kh;
        v2f a;
        a.x = rrow[k0];
        a.y = rrow[k0 + 1];
        const int i0 = t - k0;
        const int i1 = i0 - 1;
        v2f b;
        b.x = (i0 >= 0) ? ch[i0] : 0.0f;
        b.y = (i1 >= 0) ? ch[i1] : 0.0f;
        acc = __builtin_amdgcn_wmma_f32_16x16x4_f32(
                  false, a, false, b, (short)0, acc, false, false);
    }

    // D layout: vgpr r holds (m = r + 8*half, n = lane&15)
#pragma unroll
    for (int r = 0; r < 8; ++r) {
        const int bD = mtile * 16 + r + 8 * half;
        const int tD = tbase + n;
        if (tD < USEFUL)
            y[((size_t)bD * 2 + c) * USEFUL + tD] = acc[r];
    }
}

// ---------------------------------------------------------------------------
// Kernel C: windowed 64-tap STFT as two GEMMs (cos/sin) + magnitude.
// One block per (b,c). Frames (reflect pad, hann64 applied) staged in LDS.
//   re[t][f] = sum_j g[t][j]*cos(...),  im likewise; out[b,c,f,t] = |.|
// ---------------------------------------------------------------------------
__global__ __launch_bounds__(256) void stft_wmma(const float* __restrict__ y,
                                                 const float* __restrict__ basis,
                                                 float* __restrict__ out) {
    __shared__ float g[MROWS * PITCH];

    const int bc = blockIdx.x;               // b*2 + c
    const float* __restrict__ yc = y + (size_t)bc * USEFUL;

    // Build windowed frames: g[t][j] = hann64[j] * y_reflect[t*16 - 32 + j]
    for (int i = threadIdx.x; i < MROWS * 64; i += 256) {
        const int tt = i >> 6;               // frame row 0..175
        const int j  = i & 63;               // tap 0..63
        float v = 0.0f;
        if (tt < NFRAMES) {
            int src = tt * 16 - 32 + j;
            if (src < 0) src = -src;                         // reflect left
            if (src >= USEFUL) src = 2 * (USEFUL - 1) - src; // reflect right
            const float w = 0.5f * (1.0f - __cosf(6.283185307179586f * (float)j * (1.0f / 64.0f)));
            v = w * yc[src];
        }
        g[tt * PITCH + j] = v;
    }
    __syncthreads();

    const int wave = threadIdx.x >> 5;
    const int lane = threadIdx.x & 31;
    const int m    = lane & 15;
    const int half = lane >> 4;
    const int n    = lane & 15;

    const int NTILES = 11 * 17;              // t-tiles x f-tiles
    for (int tile = wave; tile < NTILES; tile += 8) {
        const int tb = (tile / 17) * 16;     // frame-tile base
        const int fb = (tile % 17) * 16;     // freq-tile base

        v8f re = {0.f, 0.f, 0.f, 0.f, 0.f, 0.f, 0.f, 0.f};
        v8f im = {0.f, 0.f, 0.f, 0.f, 0.f, 0.f, 0.f, 0.f};

#pragma unroll
        for (int kk = 0; kk < 16; ++kk) {
            const int k0 = kk * 4 + half * 2;
            v2f a;
            a.x = g[(tb + m) * PITCH + k0];
            a.y = g[(tb + m) * PITCH + k0 + 1];

            const float* __restrict__ bcp = basis + (size_t)k0 * NPADF + fb + n;
            v2f bcs; bcs.x = bcp[0];             bcs.y = bcp[NPADF];
            const float* __restrict__ bsp = bcp + (size_t)64 * NPADF;
            v2f bsn; bsn.x = bsp[0];             bsn.y = bsp[NPADF];

            re = __builtin_amdgcn_wmma_f32_16x16x4_f32(
                     false, a, false, bcs, (short)0, re, false, false);
            im = __builtin_amdgcn_wmma_f32_16x16x4_f32(
                     false, a, false, bsn, (short)0, im, false, false);
        }

#pragma unroll
        for (int r = 0; r < 8; ++r) {
            const int t = tb + r + 8 * half;
            const int f = fb + n;
            if (t < NFRAMES && f < NFREQ) {
                const float mag = sqrtf(re[r] * re[r] + im[r] * im[r]);
                out[((size_t)bc * NFREQ + f) * NFRAMES + t] = mag;
            }
        }
    }
}

// ---------------------------------------------------------------------------
// Launch
// ---------------------------------------------------------------------------
extern "C" void kernel_launch(void* const* d_in, const int* in_sizes, int n_in,
                              void* d_out, int out_size, void* d_ws, size_t ws_size,
                              hipStream_t stream) {
    (void)in_sizes; (void)n_in; (void)out_size; (void)ws_size;
    const float* rir   = (const float*)d_in[0];   // (64, 2, 44100)
    const float* chirp = (const float*)d_in[1];   // (2, 44100)
    float* out = (float*)d_out;                   // (64, 2, 257, 166)

    float* y     = (float*)d_ws;                  // (64, 2, 2646)  = 1.35 MB
    float* basis = y + (size_t)64 * 2 * USEFUL;   // 2 x 64 x 272   = 139 KB

    build_basis<<<(64 * NPADF + 255) / 256, 256, 0, stream>>>(basis);
    conv_wmma<<<NT_CONV, 256, 0, stream>>>(rir, chirp, y);
    stft_wmma<<<128, 256, 0, stream>>>(y, basis, out);
}